// GCN_30090540876084
// MI455X (gfx1250) — compile-verified
//
#include <hip/hip_runtime.h>
#include <math.h>

typedef float v2f __attribute__((ext_vector_type(2)));
typedef float v8f __attribute__((ext_vector_type(8)));

static constexpr int NN   = 50000;   // nodes (== 3125 * 16, exact WMMA row tiles)
static constexpr int NE   = 800000;  // edges
static constexpr int OUTW = 170;     // 32 emb + 10 cluster + 128 dec

// ---------------- degree / normalization ----------------

__global__ void k_init_deg(float* __restrict__ deg) {
  int i = blockIdx.x * blockDim.x + threadIdx.x;
  if (i < NN) deg[i] = 1.0f;  // self loop
}

__global__ void k_edge_deg(const int* __restrict__ col, float* __restrict__ deg) {
  int e = blockIdx.x * blockDim.x + threadIdx.x;
  if (e < NE) atomicAdd(&deg[col[e]], 1.0f);
}

__global__ void k_rsqrt(float* __restrict__ deg) {
  int i = blockIdx.x * blockDim.x + threadIdx.x;
  if (i < NN) deg[i] = rsqrtf(deg[i]);  // deg >= 1 always
}

// ---------------- FP32 WMMA GEMM: out = act(A[N,K] @ W[K,NOUT] + bias) ----------------
// One wave computes one 16x16 tile; blockDim.x = (NOUT/16)*32, grid.x = NN/16.
// ACT: 0 = none (no bias), 1 = bias+ReLU, 2 = bias+softplus
template<int K, int NOUT, int ACT>
__global__ void k_wmma_gemm(const float* __restrict__ A,
                            const float* __restrict__ W,
                            const float* __restrict__ bias,
                            float* __restrict__ out,
                            int ldout, int coloff) {
  const int lane = threadIdx.x & 31;
  const int wave = threadIdx.x >> 5;
  const int m    = lane & 15;   // row within tile (A), col within tile (B/C/D)
  const int kh   = lane >> 4;   // K-half select (A/B), M-half select (C/D)
  const int mr   = blockIdx.x * 16;  // row tile base
  const int nc   = wave * 16;        // col tile base

  v8f acc = {};
  const float* arow = A + (size_t)(mr + m) * K + kh * 2;
#pragma unroll
  for (int k0 = 0; k0 < K; k0 += 4) {
    // A fragment: lane m holds A[mr+m][k0+2*kh .. +1]  (8B-aligned pair)
    const float2 av = *(const float2*)(arow + k0);
    v2f a; a.x = av.x; a.y = av.y;
    // B fragment: lane m holds W[k0+2*kh][nc+m], W[k0+2*kh+1][nc+m]
    const float* bp = W + (size_t)(k0 + kh * 2) * NOUT + nc + m;
    v2f b; b.x = bp[0]; b.y = bp[NOUT];
    // D = A*B + C   (v_wmma_f32_16x16x4_f32)
    acc = __builtin_amdgcn_wmma_f32_16x16x4_f32(
        /*neg_a=*/false, a, /*neg_b=*/false, b,
        /*c_mod=*/(short)0, acc, /*reuse_a=*/false, /*reuse_b=*/false);
  }

  const int   n  = nc + m;
  const float bv = (ACT == 0) ? 0.0f : bias[n];
#pragma unroll
  for (int r = 0; r < 8; ++r) {
    int   mo = mr + r + kh * 8;   // C/D layout: VGPR r -> M = r + 8*(lane>=16)
    float v  = acc[r] + bv;
    if (ACT == 1) v = fmaxf(v, 0.0f);
    else if (ACT == 2) v = (v > 20.0f) ? v : log1pf(expf(v));  // softplus
    out[(size_t)mo * ldout + n + coloff] = v;
  }
}

// ---------------- GCN aggregation ----------------

// a[i,:] = h[i,:] * dinv[i]^2   (self-loop contribution, also zero-inits accumulator)
template<int F>
__global__ void k_self_init(const float* __restrict__ h, const float* __restrict__ dinv,
                            float* __restrict__ a) {
  int t = blockIdx.x * blockDim.x + threadIdx.x;
  if (t < NN * F) {
    float d = dinv[t / F];
    a[t] = h[t] * d * d;
  }
}

// per-edge scatter: a[col,:] += h[row,:] * dinv[row]*dinv[col]; 4 floats per thread
template<int F>
__global__ void k_edge_agg(const int* __restrict__ rows, const int* __restrict__ cols,
                           const float* __restrict__ dinv,
                           const float* __restrict__ h, float* __restrict__ a) {
  constexpr int CH = F / 4;
  int t = blockIdx.x * blockDim.x + threadIdx.x;
  if (t >= NE * CH) return;
  int e  = t / CH;
  int f4 = (t % CH) * 4;
  int r = rows[e], c = cols[e];
  float nrm = dinv[r] * dinv[c];
  const float4 hv = *(const float4*)(h + (size_t)r * F + f4);
  float* ap = a + (size_t)c * F + f4;
  atomicAdd(ap + 0, hv.x * nrm);
  atomicAdd(ap + 1, hv.y * nrm);
  atomicAdd(ap + 2, hv.z * nrm);
  atomicAdd(ap + 3, hv.w * nrm);
}

template<int F>
__global__ void k_bias_relu(float* __restrict__ a, const float* __restrict__ bias) {
  int t = blockIdx.x * blockDim.x + threadIdx.x;
  if (t < NN * F) a[t] = fmaxf(a[t] + bias[t % F], 0.0f);
}

// ---------------- small output kernels ----------------

__global__ void k_copy_emb(const float* __restrict__ emb, float* __restrict__ out) {
  int t = blockIdx.x * blockDim.x + threadIdx.x;
  if (t < NN * 32) out[(size_t)(t / 32) * OUTW + (t % 32)] = emb[t];
}

__global__ void k_head(const float* __restrict__ emb, const float* __restrict__ Wp,
                       const float* __restrict__ bp, float* __restrict__ out) {
  int t = blockIdx.x * blockDim.x + threadIdx.x;
  if (t >= NN * 10) return;
  int i = t / 10, j = t % 10;
  const float* er = emb + (size_t)i * 32;
  float s = bp[j];
#pragma unroll
  for (int k = 0; k < 32; ++k) s += er[k] * Wp[k * 10 + j];
  out[(size_t)i * OUTW + 32 + j] = s;
}

// ---------------- launcher ----------------

extern "C" void kernel_launch(void* const* d_in, const int* in_sizes, int n_in,
                              void* d_out, int out_size, void* d_ws, size_t ws_size,
                              hipStream_t stream) {
  (void)in_sizes; (void)n_in; (void)out_size; (void)ws_size;
  const float* X   = (const float*)d_in[0];
  const int*   ei  = (const int*)d_in[1];
  const float* W1  = (const float*)d_in[2];
  const float* b1  = (const float*)d_in[3];
  const float* W2  = (const float*)d_in[4];
  const float* b2  = (const float*)d_in[5];
  const float* Wp  = (const float*)d_in[6];
  const float* bp  = (const float*)d_in[7];
  const float* Wd1 = (const float*)d_in[8];
  const float* bd1 = (const float*)d_in[9];
  const float* Wd2 = (const float*)d_in[10];
  const float* bd2 = (const float*)d_in[11];
  const float* Wd3 = (const float*)d_in[12];
  const float* bd3 = (const float*)d_in[13];
  float* out = (float*)d_out;

  const int* rows = ei;        // edge_index[0] (source)
  const int* cols = ei + NE;   // edge_index[1] (aggregation target)

  // workspace layout (floats): 321*N total ~= 64.2 MB
  float* ws   = (float*)d_ws;
  float* dinv = ws;                    // N      (deg -> rsqrt in place)
  float* h1   = ws + (size_t)NN;       // N*64   X@W1
  float* a1   = h1 + (size_t)NN * 64;  // N*64   aggregated layer 1
  float* d1   = h1;                    // N*128  decoder h1 (overlays dead h1+a1)
  float* h2   = a1 + (size_t)NN * 64;  // N*32   a1@W2
  float* emb  = h2 + (size_t)NN * 32;  // N*32   embedding
  float* d2   = emb + (size_t)NN * 32; // N*128  decoder h2

  auto cdiv = [](int a, int b) { return (a + b - 1) / b; };

  // symmetric normalization D^-1/2
  k_init_deg<<<cdiv(NN, 256), 256, 0, stream>>>(dinv);
  k_edge_deg<<<cdiv(NE, 256), 256, 0, stream>>>(cols, dinv);
  k_rsqrt<<<cdiv(NN, 256), 256, 0, stream>>>(dinv);

  // GCN layer 1: h1 = X@W1; a1 = norm-agg(h1); relu(a1 + b1)
  k_wmma_gemm<128, 64, 0><<<NN / 16, 4 * 32, 0, stream>>>(X, W1, nullptr, h1, 64, 0);
  k_self_init<64><<<cdiv(NN * 64, 256), 256, 0, stream>>>(h1, dinv, a1);
  k_edge_agg<64><<<cdiv(NE * 16, 256), 256, 0, stream>>>(rows, cols, dinv, h1, a1);
  k_bias_relu<64><<<cdiv(NN * 64, 256), 256, 0, stream>>>(a1, b1);

  // GCN layer 2: h2 = a1@W2; emb = relu(norm-agg(h2) + b2)
  k_wmma_gemm<64, 32, 0><<<NN / 16, 2 * 32, 0, stream>>>(a1, W2, nullptr, h2, 32, 0);
  k_self_init<32><<<cdiv(NN * 32, 256), 256, 0, stream>>>(h2, dinv, emb);
  k_edge_agg<32><<<cdiv(NE * 8, 256), 256, 0, stream>>>(rows, cols, dinv, h2, emb);
  k_bias_relu<32><<<cdiv(NN * 32, 256), 256, 0, stream>>>(emb, b2);

  // outputs: emb -> cols [0,32), cluster head -> cols [32,42)
  k_copy_emb<<<cdiv(NN * 32, 256), 256, 0, stream>>>(emb, out);
  k_head<<<cdiv(NN * 10, 256), 256, 0, stream>>>(emb, Wp, bp, out);

  // decoder: relu, relu, softplus -> cols [42,170) of d_out
  k_wmma_gemm<32, 128, 1><<<NN / 16, 8 * 32, 0, stream>>>(emb, Wd1, bd1, d1, 128, 0);
  k_wmma_gemm<128, 128, 1><<<NN / 16, 8 * 32, 0, stream>>>(d1, Wd2, bd2, d2, 128, 0);
  k_wmma_gemm<128, 128, 2><<<NN / 16, 8 * 32, 0, stream>>>(d2, Wd3, bd3, out, OUTW, 42);
}